// DCGPartSeg_3521873183199
// MI455X (gfx1250) — compile-verified
//
#include <hip/hip_runtime.h>
#include <hip/hip_bf16.h>

#define NPTS 2048
#define KNN  20
#define NBATCH 4
#define SEGC 50
#define NEGS 0.2f
#define BN_SC 0.9999950000124999f   // 1/sqrt(1+1e-5)
#define FNEG_INF (-3.402823466e38f)

typedef __attribute__((ext_vector_type(2))) float v2f;
typedef __attribute__((ext_vector_type(8))) float v8f;

__device__ __forceinline__ v8f wmma_f32_k4(v2f a, v2f b, v8f c) {
  // D = A(16x4 f32) * B(4x16 f32) + C(16x16 f32)
  return __builtin_amdgcn_wmma_f32_16x16x4_f32(false, a, false, b, (short)0, c,
                                               false, false);
}

__device__ __forceinline__ float leaky(float y) { return y > 0.f ? y : NEGS * y; }

// ---------------------------------------------------------------------------
// Kernel 1: 16-row Gram stripe via WMMA into LDS, then per-row top-K indices.
// src viewed as src[(b*srcCS + srcOff + c)*NPTS + n]; Gram[n][m] = sum_c x_cn x_cm
// grid = (NPTS/16, B), block = 256 (8 waves). Dyn LDS = (16*NPTS + 16*64)*4 B.
// C compile-time: guards fold away for C%4==0; branchless clamp+mask otherwise.
// ---------------------------------------------------------------------------
template <int C>
__global__ void gram_topk_kernel(const float* src, int srcCS, int srcOff,
                                 int* idxOut) {
  constexpr int Cpad = (C + 3) & ~3;
  extern __shared__ float smem[];
  float* sG = smem;              // 16 x 2048 stripe
  float* sA = smem + 16 * NPTS;  // 16 x 64 (zero-padded A tile)
  const int tid = threadIdx.x;
  const int b = blockIdx.y;
  const int n0 = blockIdx.x * 16;
  const float* sB = src + ((size_t)b * srcCS + srcOff) * (size_t)NPTS;

  for (int i = tid; i < 16 * Cpad; i += 256) {
    int m = i / Cpad, c = i % Cpad;
    sA[m * 64 + c] = (c < C) ? sB[(size_t)c * NPTS + n0 + m] : 0.f;
  }
  __syncthreads();

  const int wave = tid >> 5, lane = tid & 31;
  const bool hi = lane >= 16;
  const int mrow = lane & 15;
  const int ksel = hi ? 2 : 0;
  for (int jt = wave; jt < NPTS / 16; jt += 8) {
    const int j0 = jt * 16;
    const int ncol = j0 + (lane & 15);
    const float* bp0 = sB + (size_t)ksel * NPTS + ncol;  // row kb, this lane's col
    v8f acc = {0.f, 0.f, 0.f, 0.f, 0.f, 0.f, 0.f, 0.f};
#pragma unroll
    for (int k0 = 0; k0 < Cpad; k0 += 4) {
      v2f a = *(const v2f*)&sA[mrow * 64 + k0 + ksel];  // 8B-aligned pair
      v2f bf;
      if constexpr ((C & 3) == 0) {
        bf.x = bp0[(size_t)k0 * NPTS];
        bf.y = bp0[(size_t)k0 * NPTS + NPTS];
      } else {
        // branchless, in-bounds: clamp row then zero-mask (guards 0*Inf=NaN)
        int kb = k0 + ksel;
        int r0 = kb < C ? kb : C - 1;
        int r1 = kb + 1 < C ? kb + 1 : C - 1;
        float m0 = kb < C ? 1.f : 0.f;
        float m1 = kb + 1 < C ? 1.f : 0.f;
        bf.x = sB[(size_t)r0 * NPTS + ncol] * m0;
        bf.y = sB[(size_t)r1 * NPTS + ncol] * m1;
      }
      acc = wmma_f32_k4(a, bf, acc);
    }
#pragma unroll
    for (int r = 0; r < 8; ++r) {
      int row = r + (hi ? 8 : 0);
      sG[row * NPTS + j0 + (lane & 15)] = acc[r];
    }
  }
  __syncthreads();

  // top-K per row (descending, smallest index wins ties == jax.lax.top_k)
  for (int rr = 0; rr < 2; ++rr) {
    const int row = wave * 2 + rr;
    float* rowp = sG + row * NPTS;
    int* outp = idxOut + ((size_t)b * NPTS + n0 + row) * KNN;
    for (int sel = 0; sel < KNN; ++sel) {
      float best = FNEG_INF;
      int bidx = NPTS;
      for (int j = lane; j < NPTS; j += 32) {
        float v = rowp[j];
        if (v > best) { best = v; bidx = j; }  // strict > keeps smallest j in-lane
      }
      for (int off = 16; off > 0; off >>= 1) {
        float ob = __shfl_down(best, off);
        int oi = __shfl_down(bidx, off);
        if (ob > best || (ob == best && oi < bidx)) { best = ob; bidx = oi; }
      }
      bidx = __shfl(bidx, 0);
      if (lane == 0) { outp[sel] = bidx; rowp[bidx] = FNEG_INF; }
    }
  }
}

// ---------------------------------------------------------------------------
// Kernel 2: fused gather + [feat-center, feat] + conv2d(W1)+BN+leaky
//           (+ optional conv2d(W2)+BN+leaky) + max over K  -> one (B,64,N) slab.
// grid = (NPTS, B), block = 256. W2 == nullptr -> single-conv (stage C).
// ---------------------------------------------------------------------------
__global__ void edge_conv_kernel(const float* src, int srcCS, int srcOff, int C,
                                 const int* idx, const float* W1, const float* g1,
                                 const float* b1, const float* W2, const float* g2,
                                 const float* b2, float* dst, int dstCS, int dstOff) {
  __shared__ float sW1[64 * 128];
  __shared__ float sFin[128 * KNN];
  __shared__ float sH[64 * KNN];
  __shared__ float sCtr[64];
  __shared__ int sIdx[KNN];
  __shared__ float sRed[4 * 64];
  const int tid = threadIdx.x;
  const int b = blockIdx.y, n = blockIdx.x;
  const int twoC = 2 * C;
  const float* sB = src + ((size_t)b * srcCS + srcOff) * (size_t)NPTS;

  if (tid < C) sCtr[tid] = sB[(size_t)tid * NPTS + n];
  if (tid >= 64 && tid < 64 + KNN)
    sIdx[tid - 64] = idx[((size_t)b * NPTS + n) * KNN + (tid - 64)];
  for (int i = tid; i < 64 * twoC; i += 256) sW1[i] = W1[i];
  __syncthreads();

  for (int i = tid; i < KNN * C; i += 256) {
    int k = i / C, c = i % C;
    float f = sB[(size_t)c * NPTS + sIdx[k]];
    sFin[c * KNN + k] = f - sCtr[c];   // channels [0,C): feat - center
    sFin[(C + c) * KNN + k] = f;       // channels [C,2C): feat
  }
  __syncthreads();

  for (int i = tid; i < 64 * KNN; i += 256) {
    int o = i / KNN, k = i % KNN;
    float s = 0.f;
    const float* wr = &sW1[o * twoC];
    for (int c = 0; c < twoC; ++c) s += wr[c] * sFin[c * KNN + k];
    s *= BN_SC; s = s * g1[o] + b1[o];
    sH[i] = leaky(s);
  }
  __syncthreads();

  const int o = tid & 63, kg = tid >> 6;
  float m = FNEG_INF;
  for (int k = kg; k < KNN; k += 4) {
    float v;
    if (W2) {
      float s = 0.f;
      const float* wr = &W2[(size_t)o * 64];
      for (int c = 0; c < 64; ++c) s += wr[c] * sH[c * KNN + k];
      s *= BN_SC; s = s * g2[o] + b2[o];
      v = leaky(s);
    } else {
      v = sH[o * KNN + k];
    }
    m = fmaxf(m, v);
  }
  sRed[kg * 64 + o] = m;
  __syncthreads();
  if (kg == 0) {
    m = fmaxf(fmaxf(sRed[o], sRed[64 + o]), fmaxf(sRed[128 + o], sRed[192 + o]));
    dst[((size_t)b * dstCS + dstOff + o) * (size_t)NPTS + n] = m;
  }
}

// ---------------------------------------------------------------------------
// Kernel 3: generic WMMA f32 GEMM  out[b,o,n] = act(BN*g*(W[o,:]·in[b,:,n]+rowAdd)+b)
// grid = (NPTS/128, ceil(O/16), B), block = 256 (8 waves x one 16x16 tile each).
// Dyn LDS = 16*C*4 + 1024 bytes (W tile + async-copy scratch).
// Demonstrates the CDNA5 ASYNCcnt path: global_load_async_to_lds_b32 stages one
// cacheline of the B-stream into scratch LDS (never read -> harmless prefetch).
// ---------------------------------------------------------------------------
template <int C>
__global__ void gemm_bn_wmma_kernel(const float* in, int inCS, int inOff,
                                    const float* W, int ldW, int wOff,
                                    const float* gscale, const float* bias,
                                    const float* rowAdd, float* out, int outCS,
                                    int O, int act) {
  extern __shared__ float smem[];
  float* sW = smem;              // 16 x C
  float* sScr = smem + 16 * C;   // 256 floats async scratch
  const int tid = threadIdx.x;
  const int b = blockIdx.z;
  const int o0 = blockIdx.y * 16;
  const int nb = blockIdx.x * 128;
  const float* inB = in + ((size_t)b * inCS + inOff) * (size_t)NPTS;

  // Async LDS-staging prefetch of the first B rows (ASYNCcnt-tracked).
  {
    unsigned lds_dst = (unsigned)(size_t)&sScr[tid];
    const float* gsrc = inB + nb + (tid & 127);
    asm volatile("global_load_async_to_lds_b32 %0, %1, off"
                 :: "v"(lds_dst), "v"(gsrc) : "memory");
  }

  for (int i = tid; i < 16 * C; i += 256) {
    int m = i / C, c = i % C;
    int o = o0 + m;
    sW[m * C + c] = (o < O) ? W[(size_t)o * ldW + wOff + c] : 0.f;
  }
  asm volatile("s_wait_asynccnt 0x0" ::: "memory");
  __syncthreads();

  const int wave = tid >> 5, lane = tid & 31;
  const bool hi = lane >= 16;
  const int mrow = lane & 15;
  const int ksel = hi ? 2 : 0;
  const int ncol = nb + wave * 16 + (lane & 15);
  const float* bp0 = inB + (size_t)ksel * NPTS + ncol;
  v8f acc = {0.f, 0.f, 0.f, 0.f, 0.f, 0.f, 0.f, 0.f};
#pragma unroll
  for (int k0 = 0; k0 < C; k0 += 4) {  // C multiple of 4 for all instantiations
    v2f a = *(const v2f*)&sW[mrow * C + k0 + ksel];  // 8B-aligned pair
    v2f bf;
    bf.x = bp0[(size_t)k0 * NPTS];
    bf.y = bp0[(size_t)k0 * NPTS + NPTS];
    acc = wmma_f32_k4(a, bf, acc);
  }
#pragma unroll
  for (int r = 0; r < 8; ++r) {
    int o = o0 + r + (hi ? 8 : 0);
    if (o < O) {
      float y = acc[r];
      if (rowAdd) y += rowAdd[(size_t)b * O + o];
      if (gscale) { y *= BN_SC; y = y * gscale[o] + bias[o]; }
      if (act) y = leaky(y);
      out[((size_t)b * outCS + o) * (size_t)NPTS + ncol] = y;
    }
  }
}

// Kernel 4: gmax[b,o] = max_n y[b,o,n].   grid=(Cn,B), block=256
__global__ void rowmax_kernel(const float* in, float* outv, int Cn) {
  __shared__ float red[256];
  const int b = blockIdx.y, o = blockIdx.x, tid = threadIdx.x;
  const float* r = in + ((size_t)b * Cn + o) * (size_t)NPTS;
  float m = FNEG_INF;
  for (int n = tid; n < NPTS; n += 256) m = fmaxf(m, r[n]);
  red[tid] = m;
  __syncthreads();
  for (int s = 128; s > 0; s >>= 1) {
    if (tid < s) red[tid] = fmaxf(red[tid], red[tid + s]);
    __syncthreads();
  }
  if (tid == 0) outv[(size_t)b * Cn + o] = red[0];
}

// Kernel 5: lc[b,o] = leaky(BN*g7*(W7[o,:]·l[b,:]) + b7).  grid=B, block=64
__global__ void lc_kernel(const float* l, const float* W7, const float* g7,
                          const float* b7, float* lcv) {
  const int b = blockIdx.x, o = threadIdx.x;
  float s = 0.f;
  for (int c = 0; c < 16; ++c) s += W7[o * 16 + c] * l[b * 16 + c];
  s *= BN_SC; s = s * g7[o] + b7[o];
  lcv[b * 64 + o] = leaky(s);
}

// Kernel 6: hoist W8's broadcast channels: base8[b,o] = W8[o,0:1024]·gmax + W8[o,1024:1088]·lc
__global__ void base8_kernel(const float* W8, const float* gmax, const float* lcv,
                             float* base8) {
  const int b = blockIdx.x, o = threadIdx.x;  // 256 threads
  const float* wr = W8 + (size_t)o * 1280;
  float s = 0.f;
  for (int c = 0; c < 1024; ++c) s += wr[c] * gmax[b * 1024 + c];
  for (int j = 0; j < 64; ++j) s += wr[1024 + j] * lcv[b * 64 + j];
  base8[b * 256 + o] = s;
}

extern "C" void kernel_launch(void* const* d_in, const int* in_sizes, int n_in,
                              void* d_out, int out_size, void* d_ws, size_t ws_size,
                              hipStream_t stream) {
  const float* x  = (const float*)d_in[0];
  const float* l  = (const float*)d_in[1];
  const float* W1 = (const float*)d_in[2];
  const float* W2 = (const float*)d_in[3];
  const float* W3 = (const float*)d_in[4];
  const float* W4 = (const float*)d_in[5];
  const float* W5 = (const float*)d_in[6];
  const float* W6 = (const float*)d_in[7];
  const float* W7 = (const float*)d_in[8];
  const float* W8 = (const float*)d_in[9];
  const float* W9 = (const float*)d_in[10];
  const float* W10 = (const float*)d_in[11];
  const float* W11 = (const float*)d_in[12];
  const float *g1 = (const float*)d_in[13], *b1 = (const float*)d_in[14];
  const float *g2 = (const float*)d_in[15], *b2 = (const float*)d_in[16];
  const float *g3 = (const float*)d_in[17], *b3 = (const float*)d_in[18];
  const float *g4 = (const float*)d_in[19], *b4 = (const float*)d_in[20];
  const float *g5 = (const float*)d_in[21], *b5 = (const float*)d_in[22];
  const float *g6 = (const float*)d_in[23], *b6 = (const float*)d_in[24];
  const float *g7 = (const float*)d_in[25], *b7 = (const float*)d_in[26];
  const float *g8 = (const float*)d_in[27], *b8 = (const float*)d_in[28];
  const float *g9 = (const float*)d_in[29], *b9 = (const float*)d_in[30];
  const float *g10 = (const float*)d_in[31], *b10 = (const float*)d_in[32];

  char* ws = (char*)d_ws;
  size_t off = 0;
  int* idxp = (int*)(ws + off);      off += (size_t)NBATCH * NPTS * KNN * 4;   // 640KB
  float* x123 = (float*)(ws + off);  off += (size_t)NBATCH * 192 * NPTS * 4;   // 6MB
  float* y6 = (float*)(ws + off);    off += (size_t)NBATCH * 1024 * NPTS * 4;  // 32MB
  float* gmax = (float*)(ws + off);  off += (size_t)NBATCH * 1024 * 4;
  float* lcv = (float*)(ws + off);   off += (size_t)NBATCH * 64 * 4;
  float* base8 = (float*)(ws + off); off += (size_t)NBATCH * 256 * 4;
  float* h8 = (float*)(ws + off);    off += (size_t)NBATCH * 256 * NPTS * 4;   // 8MB
  float* h9 = (float*)(ws + off);    off += (size_t)NBATCH * 256 * NPTS * 4;   // 8MB
  float* h10 = (float*)(ws + off);   off += (size_t)NBATCH * 128 * NPTS * 4;   // 4MB

  const dim3 gGram(NPTS / 16, NBATCH);
  const dim3 gEdge(NPTS, NBATCH);
  const size_t ldsGram = (16 * NPTS + 16 * 64) * sizeof(float);

  // Stage A: knn on x (C=3), edge conv W1->W2 -> x1 (x123 ch 0..63)
  gram_topk_kernel<3><<<gGram, 256, ldsGram, stream>>>(x, 3, 0, idxp);
  edge_conv_kernel<<<gEdge, 256, 0, stream>>>(x, 3, 0, 3, idxp, W1, g1, b1, W2, g2,
                                              b2, x123, 192, 0);
  // Stage B: knn on x1, edge conv W3->W4 -> x2 (ch 64..127)
  gram_topk_kernel<64><<<gGram, 256, ldsGram, stream>>>(x123, 192, 0, idxp);
  edge_conv_kernel<<<gEdge, 256, 0, stream>>>(x123, 192, 0, 64, idxp, W3, g3, b3,
                                              W4, g4, b4, x123, 192, 64);
  // Stage C: knn on x2, single edge conv W5 -> x3 (ch 128..191)
  gram_topk_kernel<64><<<gGram, 256, ldsGram, stream>>>(x123, 192, 64, idxp);
  edge_conv_kernel<<<gEdge, 256, 0, stream>>>(x123, 192, 64, 64, idxp, W5, g5, b5,
                                              nullptr, nullptr, nullptr, x123, 192, 128);
  // W6: (B,192,N) -> (B,1024,N), then global max over n
  gemm_bn_wmma_kernel<192><<<dim3(NPTS / 128, 64, NBATCH), 256, 16 * 192 * 4 + 1024,
                             stream>>>(x123, 192, 0, W6, 192, 0, g6, b6, nullptr, y6,
                                       1024, 1024, 1);
  rowmax_kernel<<<dim3(1024, NBATCH), 256, 0, stream>>>(y6, gmax, 1024);
  lc_kernel<<<NBATCH, 64, 0, stream>>>(l, W7, g7, b7, lcv);
  base8_kernel<<<NBATCH, 256, 0, stream>>>(W8, gmax, lcv, base8);
  // W8: broadcast channels hoisted into base8; GEMM over last 192 channels
  gemm_bn_wmma_kernel<192><<<dim3(NPTS / 128, 16, NBATCH), 256, 16 * 192 * 4 + 1024,
                             stream>>>(x123, 192, 0, W8, 1280, 1088, g8, b8, base8,
                                       h8, 256, 256, 1);
  gemm_bn_wmma_kernel<256><<<dim3(NPTS / 128, 16, NBATCH), 256, 16 * 256 * 4 + 1024,
                             stream>>>(h8, 256, 0, W9, 256, 0, g9, b9, nullptr, h9,
                                       256, 256, 1);
  gemm_bn_wmma_kernel<256><<<dim3(NPTS / 128, 8, NBATCH), 256, 16 * 256 * 4 + 1024,
                             stream>>>(h9, 256, 0, W10, 256, 0, g10, b10, nullptr,
                                       h10, 128, 128, 1);
  // W11: plain einsum, no BN/act, O=50 -> d_out (B,50,N)
  gemm_bn_wmma_kernel<128><<<dim3(NPTS / 128, 4, NBATCH), 256, 16 * 128 * 4 + 1024,
                             stream>>>(h10, 128, 0, W11, 128, 0, nullptr, nullptr,
                                       nullptr, (float*)d_out, SEGC, SEGC, 0);
}